// Sampler_45509473468599
// MI455X (gfx1250) — compile-verified
//
#include <hip/hip_runtime.h>

// ---------------------------------------------------------------------------
// Sampler: temperature-scaled softmax probs + token sampling.
// B=256 rows, V=128000 cols, fp32.
// d_out layout (float): [0..255] tokens (as float), [256 ..] probs row-major.
// d_ws: per-row stats {rowmax2(base-2), 1/sumexp, scl(=invT*log2e), pad}.
// ---------------------------------------------------------------------------

#define B_ROWS   256
#define V_COLS   128000
#define T1       512                  // pass-1 block: 16 waves
#define T2       256                  // pass-2 block
#define CHUNK    4096                 // floats per DMA chunk (16 KB)
#define NCHUNK   ((V_COLS + CHUNK - 1) / CHUNK)   // 32 (last chunk = 1024)
#define NWAVES   (T1 / 32)            // 16

#define LOG2E    1.4426950408889634f

typedef unsigned int u32x4 __attribute__((ext_vector_type(4)));
typedef int          i32x4 __attribute__((ext_vector_type(4)));
typedef int          i32x8 __attribute__((ext_vector_type(8)));
typedef float        f32x4 __attribute__((ext_vector_type(4)));

// ---- Tensor Data Mover: build D# (CDNA5 ISA) and issue ----
// Group0: [1:0]=count=1, [63:32]=lds_addr, [120:64]=global_addr, [127:126]=type=2
// Group1: [17:16]=data_size(2 -> 4B), [79:48]=tensor_dim0, [111:80]=tensor_dim1,
//         [127:112]=tile_dim0, [207:160]=tensor_dim0_stride
__device__ __forceinline__ void tdm_load_1d(const float* gptr, unsigned lds_byte_off,
                                            int nelem) {
  unsigned long long ga = (unsigned long long)(size_t)gptr;
  u32x4 g0;
  g0.x = 1u;                                             // count=1, user mode
  g0.y = lds_byte_off;                                   // lds_addr (bytes)
  g0.z = (unsigned)(ga & 0xFFFFFFFFull);                 // global_addr[31:0]
  g0.w = (unsigned)((ga >> 32) & 0x1FFFFFFull) | (2u << 30); // addr[56:32] | type=2
  i32x8 g1;
  g1[0] = (int)(2u << 16);                               // data_size = 4 bytes
  g1[1] = (int)(((unsigned)nelem & 0xFFFFu) << 16);      // tensor_dim0[15:0]
  g1[2] = (int)((((unsigned)nelem >> 16) & 0xFFFFu) | (1u << 16)); // dim0 hi | dim1=1
  g1[3] = (int)(((unsigned)nelem & 0xFFFFu) << 16);      // tile_dim0 = nelem
  g1[4] = 0;                                             // tile_dim1/2 unused
  g1[5] = nelem;                                         // tensor_dim0_stride lo
  g1[6] = 0;
  g1[7] = 0;
  i32x4 z4 = {0, 0, 0, 0};
#if defined(__clang_major__) && (__clang_major__ >= 23)
  i32x8 z8 = {0, 0, 0, 0, 0, 0, 0, 0};
  __builtin_amdgcn_tensor_load_to_lds(g0, g1, z4, z4, z8, 0);
#else
  __builtin_amdgcn_tensor_load_to_lds(g0, g1, z4, z4, 0);
#endif
}

// Counter-based hash RNG -> uniform (0,1). Deterministic per (row,col).
__device__ __forceinline__ float hash_uniform(unsigned row, unsigned col) {
  unsigned h = row * 0x9E3779B1u + col * 0x85EBCA77u + 42u;
  h ^= h >> 16; h *= 0x7FEB352Du;
  h ^= h >> 15; h *= 0x846CA68Bu;
  h ^= h >> 16;
  return ((h >> 8) + 1u) * (1.0f / 16777218.0f);   // strictly in (0,1)
}

// ---------------------------------------------------------------------------
// Pass 1: per-row online softmax (base-2) + greedy argmax + Gumbel-max sample.
// One block (16 waves) per row; wave 0 drives double-buffered TDM loads.
// ---------------------------------------------------------------------------
__global__ __launch_bounds__(T1)
void sampler_pass1(const float* __restrict__ logits,
                   const float* __restrict__ temps,
                   float* __restrict__ out_tokens,     // d_out[0..B)
                   float* __restrict__ rowstats) {     // d_ws, 4 floats/row
  extern __shared__ float lds[];                       // 2*CHUNK floats + scratch
  const int row = blockIdx.x;
  const int tid = threadIdx.x;
  const float t = temps[row];
  const bool greedy = (t == 0.0f);
  const float invT = greedy ? 1.0f : (1.0f / t);
  const float scl  = invT * LOG2E;                     // base-2 scaling
  const float* rowp = logits + (size_t)row * V_COLS;

  // Prime the pipeline: wave 0 issues chunks 0 and 1.
  if (tid < 32) {
    tdm_load_1d(rowp, 0u, CHUNK);
    int n1 = (V_COLS - CHUNK) < CHUNK ? (V_COLS - CHUNK) : CHUNK;
    tdm_load_1d(rowp + CHUNK, (unsigned)(CHUNK * 4), n1);
  }

  float m = -3.402823466e38f;                // running max of s2 (base-2 scaled)
  float ssum = 0.0f;                         // sum of 2^(s2 - m)
  int   midx = 0;                            // argmax index (greedy token)
  float kmax = -3.402823466e38f; int kidx = 0;   // Gumbel-max key argmax

  for (int c = 0; c < NCHUNK; ++c) {
    if (tid < 32) {
      if (c + 1 < NCHUNK) __builtin_amdgcn_s_wait_tensorcnt(1); // chunk c landed
      else                __builtin_amdgcn_s_wait_tensorcnt(0);
    }
    __syncthreads();

    const f32x4* buf = (const f32x4*)(lds + (c & 1) * CHUNK);
    const int base = c * CHUNK;
#pragma unroll
    for (int j = 0; j < CHUNK / 4 / T1; ++j) {           // 2 iterations
      const int v4 = j * T1 + tid;
      const f32x4 x4 = buf[v4];
      const int col0 = base + v4 * 4;
      const float xs[4] = {x4.x, x4.y, x4.z, x4.w};
#pragma unroll
      for (int k = 0; k < 4; ++k) {
        const int col = col0 + k;
        if (col < V_COLS) {
          const float s2 = xs[k] * scl;
          // Branchy online softmax: common path costs one v_exp_f32.
          if (s2 <= m) {
            ssum += exp2f(s2 - m);
          } else {
            ssum = ssum * exp2f(m - s2) + 1.0f;
            m = s2; midx = col;
          }
          if (!greedy) {
            // Gumbel-max in base-2: key = s2 - log2(-log2 u)  (const shift dropped)
            const float u = hash_uniform((unsigned)row, (unsigned)col);
            const float g = s2 - __log2f(-__log2f(u));
            if (g > kmax) { kmax = g; kidx = col; }
          }
        }
      }
    }
    __syncthreads();   // all waves done with buf[c&1] before it is reloaded

    if (tid < 32 && (c + 2) < NCHUNK) {
      const int base2 = (c + 2) * CHUNK;
      int n = V_COLS - base2; if (n > CHUNK) n = CHUNK;
      tdm_load_1d(rowp + base2, (unsigned)((c & 1) * CHUNK * 4), n);
    }
  }

  // ---- wave32 shuffle reduction ----
#pragma unroll
  for (int off = 16; off > 0; off >>= 1) {
    float mo = __shfl_down(m, off);
    float so = __shfl_down(ssum, off);
    int   io = __shfl_down(midx, off);
    if (mo > m || (mo == m && io < midx)) {
      ssum = ssum * exp2f(m - mo) + so;
      m = mo; midx = io;
    } else {
      ssum += so * exp2f(mo - m);
    }
    float ko = __shfl_down(kmax, off); int kio = __shfl_down(kidx, off);
    if (ko > kmax || (ko == kmax && kio < kidx)) { kmax = ko; kidx = kio; }
  }

  // ---- cross-wave reduction through LDS scratch (after the two buffers) ----
  float* sc  = lds + 2 * CHUNK;
  int*   sci = (int*)sc;
  const int wid = tid >> 5;
  if ((tid & 31) == 0) {
    sc [0 * NWAVES + wid] = m;
    sci[1 * NWAVES + wid] = midx;
    sc [2 * NWAVES + wid] = ssum;
    sc [3 * NWAVES + wid] = kmax;
    sci[4 * NWAVES + wid] = kidx;
  }
  __syncthreads();
  if (tid == 0) {
    float M = sc[0]; int MI = sci[NWAVES]; float S = sc[2 * NWAVES];
    float KM = sc[3 * NWAVES]; int KI = sci[4 * NWAVES];
#pragma unroll
    for (int w = 1; w < NWAVES; ++w) {
      float mw = sc[w]; int iw = sci[NWAVES + w]; float sw = sc[2 * NWAVES + w];
      if (mw > M || (mw == M && iw < MI)) {
        S = S * exp2f(M - mw) + sw;
        M = mw; MI = iw;
      } else {
        S += sw * exp2f(mw - M);
      }
      float kw = sc[3 * NWAVES + w]; int kiw = sci[4 * NWAVES + w];
      if (kw > KM || (kw == KM && kiw < KI)) { KM = kw; KI = kiw; }
    }
    const int token = greedy ? MI : KI;
    out_tokens[row] = (float)token;
    f32x4 st; st.x = M; st.y = 1.0f / S; st.z = scl; st.w = 0.0f;
    ((f32x4*)rowstats)[row] = st;
  }
}

// ---------------------------------------------------------------------------
// Pass 2: probs = 2^(x*scl - m) * (1/sum). Logits reread hits the 192MB L2;
// probs streamed out with non-temporal 128-bit stores so they don't evict it.
// ---------------------------------------------------------------------------
__global__ __launch_bounds__(T2)
void sampler_pass2(const float* __restrict__ logits,
                   const float* __restrict__ rowstats,
                   float* __restrict__ probs) {
  const int row = blockIdx.y;
  const int i4  = blockIdx.x * T2 + threadIdx.x;     // float4 index, V/4=32000
  const f32x4 st = ((const f32x4*)rowstats)[row];
  const float m = st.x, rinv = st.y, scl = st.z;
  const f32x4* src = (const f32x4*)(logits + (size_t)row * V_COLS);
  const f32x4 x = src[i4];
  f32x4 p;
  p.x = exp2f(x.x * scl - m) * rinv;
  p.y = exp2f(x.y * scl - m) * rinv;
  p.z = exp2f(x.z * scl - m) * rinv;
  p.w = exp2f(x.w * scl - m) * rinv;
  f32x4* dst = (f32x4*)(probs + (size_t)row * V_COLS) + i4;
  __builtin_nontemporal_store(p, dst);
}

// ---------------------------------------------------------------------------
extern "C" void kernel_launch(void* const* d_in, const int* in_sizes, int n_in,
                              void* d_out, int out_size, void* d_ws, size_t ws_size,
                              hipStream_t stream) {
  const float* logits = (const float*)d_in[0];   // [256, 128000] fp32
  const float* temps  = (const float*)d_in[1];   // [256] fp32
  float* out      = (float*)d_out;               // tokens[256] ++ probs[256*128000]
  float* rowstats = (float*)d_ws;                // 256 * 4 floats

  const size_t shmem = (size_t)(2 * CHUNK + 128) * sizeof(float);  // ~33 KB
  sampler_pass1<<<dim3(B_ROWS), dim3(T1), shmem, stream>>>(
      logits, temps, out, rowstats);
  sampler_pass2<<<dim3(V_COLS / (T2 * 4), B_ROWS), dim3(T2), 0, stream>>>(
      logits, rowstats, out + B_ROWS);
}